// CausalSelfAttention_48430051230386
// MI455X (gfx1250) — compile-verified
//
#include <hip/hip_runtime.h>

#define B_   2
#define T_   2048
#define D_   1024
#define NH_  16
#define NKV_ 4
#define HD_  64
#define E_   1536   /* (NH + 2*NKV) * HD */

typedef __attribute__((ext_vector_type(2))) float v2f;
typedef __attribute__((ext_vector_type(8))) float v8f;
typedef unsigned int u32x4 __attribute__((ext_vector_type(4)));
typedef int          i32x8 __attribute__((ext_vector_type(8)));
typedef int          i32x4 __attribute__((ext_vector_type(4)));

// D = A(16x4,f32) * B(4x16,f32) + C(16x16,f32)  -- fp32 WMMA
__device__ __forceinline__ v8f wmma4(v2f a, v2f b, v8f c) {
  return __builtin_amdgcn_wmma_f32_16x16x4_f32(
      /*neg_a=*/false, a, /*neg_b=*/false, b,
      /*c_mod=*/(short)0, c, /*reuse_a=*/false, /*reuse_b=*/false);
}

// ---------------------------------------------------------------------------
// Tensor Data Mover: async 2D tile (f32 elements) global -> LDS.
//   tile_d0  : elements per row       tile_d1 : rows
//   stride0  : row stride in elements (tensor_dim0_stride)
//   pad_code : LDS pad_interval code (4 = pad after 32 DWORDs, 5 = after 64),
//              pad_amount = 1 DWORD  -> reproduces the [..][d0+1] padded arrays
// Descriptor packing per CDNA5 ISA.
// ---------------------------------------------------------------------------
__device__ __forceinline__ void tdm_load_2d_f32(unsigned lds_off, const void* gptr,
                                                unsigned tile_d0, unsigned tile_d1,
                                                unsigned stride0, unsigned pad_code) {
  const unsigned long long ga = (unsigned long long)(uintptr_t)gptr;
  u32x4 g0;
  g0[0] = 1u;                                              // count=1, user mode
  g0[1] = lds_off;                                         // lds_addr (bytes)
  g0[2] = (unsigned)(ga & 0xFFFFFFFFu);                    // global_addr[31:0]
  g0[3] = (unsigned)((ga >> 32) & 0x01FFFFFFu) | (2u << 30); // addr[56:32] | type=2
  i32x8 g1;
  g1[0] = (int)((2u << 16)                                 // data_size = 4B
                | (1u << 20)                               // pad_enable
                | (pad_code << 22));                       // pad_interval, pad_amount=0 (1 DW)
  g1[1] = (int)((tile_d0 & 0xFFFFu) << 16);                // tensor_dim0[15:0]
  g1[2] = (int)(((tile_d0 >> 16) & 0xFFFFu) |
                ((tile_d1 & 0xFFFFu) << 16));              // tensor_dim0[31:16] | tensor_dim1[15:0]
  g1[3] = (int)(((tile_d1 >> 16) & 0xFFFFu) |
                ((tile_d0 & 0xFFFFu) << 16));              // tensor_dim1[31:16] | tile_dim0
  g1[4] = (int)(tile_d1 & 0xFFFFu);                        // tile_dim1 | tile_dim2=0
  g1[5] = (int)stride0;                                    // tensor_dim0_stride[31:0]
  g1[6] = 0;                                               // stride0 hi | stride1 lo
  g1[7] = 0;                                               // stride1 hi
  const i32x4 z4 = {0, 0, 0, 0};
#if defined(__clang_major__) && (__clang_major__ >= 23)
  const i32x8 z8 = {0, 0, 0, 0, 0, 0, 0, 0};
  __builtin_amdgcn_tensor_load_to_lds(g0, g1, z4, z4, z8, 0);
#else
  __builtin_amdgcn_tensor_load_to_lds(g0, g1, z4, z4, 0);
#endif
}

__device__ __forceinline__ unsigned lds_offset(const void* p) {
  // LDS aperture occupies addr[63:32]; low 32 bits are the workgroup LDS offset.
  return (unsigned)(uintptr_t)p;
}

// ---------------------------------------------------------------------------
// OUT[m][n] = sum_k X[m][k] * W[n][k]      (X: MxK row-major, W: NxK row-major)
// Tile: 128(M) x 64(N), KC = 32. 256 threads = 8 waves in a 4x2 grid,
// each wave computes a 32x32 patch = 2x2 fp32-WMMA accumulators.
// A/B tiles staged via TDM async tensor loads (TENSORcnt).
// ---------------------------------------------------------------------------
__global__ __launch_bounds__(256)
void gemm_xwT(const float* __restrict__ X, const float* __restrict__ W,
              float* __restrict__ OUT, int M, int N, int K) {
  __shared__ float As[128][33];
  __shared__ float Bs[64][33];

  const int tid  = threadIdx.x;
  const int wid  = tid >> 5;
  const int lane = tid & 31;
  const int half = lane >> 4;
  const int r    = lane & 15;
  const int wm   = wid & 3;      // wave row  (0..3) -> 32 rows each
  const int wn   = wid >> 2;     // wave col  (0..1) -> 32 cols each
  const int M0   = blockIdx.y * 128;
  const int N0   = blockIdx.x * 64;

  const unsigned ldsA = lds_offset(&As[0][0]);
  const unsigned ldsB = lds_offset(&Bs[0][0]);

  v8f acc[2][2] = {};

  for (int k0 = 0; k0 < K; k0 += 32) {
    // ---- TDM: stage A (128x32) and B (64x32) tiles into padded LDS ----
    if (wid == 0) {
      tdm_load_2d_f32(ldsA, X + (size_t)M0 * K + k0, 32, 128, (unsigned)K, 4);
      __builtin_amdgcn_s_wait_tensorcnt(0);
    } else if (wid == 1) {
      tdm_load_2d_f32(ldsB, W + (size_t)N0 * K + k0, 32, 64, (unsigned)K, 4);
      __builtin_amdgcn_s_wait_tensorcnt(0);
    }
    __syncthreads();

#pragma unroll
    for (int kk = 0; kk < 32; kk += 4) {
      const int ka = kk + 2 * half;
      v2f a0 = { As[wm * 32 +      r][ka], As[wm * 32 +      r][ka + 1] };
      v2f a1 = { As[wm * 32 + 16 + r][ka], As[wm * 32 + 16 + r][ka + 1] };
      v2f b0 = { Bs[wn * 32 +      r][ka], Bs[wn * 32 +      r][ka + 1] };
      v2f b1 = { Bs[wn * 32 + 16 + r][ka], Bs[wn * 32 + 16 + r][ka + 1] };
      acc[0][0] = wmma4(a0, b0, acc[0][0]);
      acc[0][1] = wmma4(a0, b1, acc[0][1]);
      acc[1][0] = wmma4(a1, b0, acc[1][0]);
      acc[1][1] = wmma4(a1, b1, acc[1][1]);
    }
    __syncthreads();
  }

#pragma unroll
  for (int mi = 0; mi < 2; ++mi)
#pragma unroll
    for (int ni = 0; ni < 2; ++ni)
#pragma unroll
      for (int v = 0; v < 8; ++v) {
        const int row = M0 + wm * 32 + mi * 16 + half * 8 + v;
        const int col = N0 + wn * 32 + ni * 16 + r;
        OUT[(size_t)row * N + col] = acc[mi][ni][v];
      }
}

// ---------------------------------------------------------------------------
// RoPE + scale-folding + value-residual mix.
//   qs = rope(q) * qk_scale[h] / HD      (B,NH,T,HD)
//   ks = rope(k)                         (B,NKV,T,HD)
//   vm = (1-sig(a))*v[kv(h)] + sig(a)*v0 (B,NH,T,HD)
// ---------------------------------------------------------------------------
__global__ __launch_bounds__(256)
void rope_prep(const float* __restrict__ qkv, const float* __restrict__ cosp,
               const float* __restrict__ sinp, const float* __restrict__ v0,
               const float* __restrict__ alpha, const float* __restrict__ qscale,
               float* __restrict__ qs, float* __restrict__ ks,
               float* __restrict__ vm) {
  const int bt = blockIdx.x;
  const int b  = bt / T_;
  const int t  = bt % T_;
  const size_t rowbase = (size_t)bt * E_;

  // q and k heads with RoPE
  for (int e = threadIdx.x; e < (NH_ + NKV_) * HD_; e += blockDim.x) {
    const int d    = e & 63;
    const size_t hb = rowbase + (size_t)(e - d);
    float val = qkv[rowbase + e];
    if (d < 32) {
      const int j   = d & 15;
      const float c = cosp[t * 16 + j];
      const float s = sinp[t * 16 + j];
      const float x1 = qkv[hb + j];
      const float x2 = qkv[hb + 16 + j];
      val = (d < 16) ? (x1 * c - x2 * s) : (x1 * s + x2 * c);
    }
    if (e < NH_ * HD_) {
      const int h = e >> 6;
      val *= qscale[h] * (1.0f / (float)HD_);
      qs[((size_t)(b * NH_ + h) * T_ + t) * HD_ + d] = val;
    } else {
      const int kvh = (e - NH_ * HD_) >> 6;
      ks[((size_t)(b * NKV_ + kvh) * T_ + t) * HD_ + d] = val;
    }
  }

  // per-NH-head mixed V
  for (int i = threadIdx.x; i < NH_ * HD_; i += blockDim.x) {
    const int h   = i >> 6;
    const int d   = i & 63;
    const int kvh = h >> 2;              // NH/NKV = 4
    const float a  = 1.0f / (1.0f + __expf(-alpha[h]));
    const float vv = qkv[rowbase + (size_t)(NH_ + NKV_) * HD_ + kvh * HD_ + d];
    const size_t oi = ((size_t)(b * NH_ + h) * T_ + t) * HD_ + d;
    vm[oi] = (1.0f - a) * vv + a * v0[oi];
  }
}

// ---------------------------------------------------------------------------
// Causal flash attention. grid = (T/128, NH, B), block = 128 (4 waves).
// Each wave owns 32 query rows; K/V 32x64 tiles staged in LDS via TDM.
// ---------------------------------------------------------------------------
__global__ __launch_bounds__(128)
void attn(const float* __restrict__ qs, const float* __restrict__ ks,
          const float* __restrict__ vm, float* __restrict__ y) {
  __shared__ float Ks[32][65];
  __shared__ float Vs[32][65];
  __shared__ float Pb[4][32][33];

  const int tid  = threadIdx.x;
  const int wid  = tid >> 5;
  const int lane = tid & 31;
  const int half = lane >> 4;
  const int r    = lane & 15;

  const int b   = blockIdx.z;
  const int h   = blockIdx.y;
  const int qt  = blockIdx.x;
  const int kvh = h >> 2;
  const int q0  = qt * 128 + wid * 32;

  const float* Q  = qs + ((size_t)(b * NH_  + h)   * T_ + q0) * HD_;
  const float* Kp = ks +  (size_t)(b * NKV_ + kvh) * T_ * HD_;
  const float* Vp = vm +  (size_t)(b * NH_  + h)   * T_ * HD_;

  const unsigned ldsK = lds_offset(&Ks[0][0]);
  const unsigned ldsV = lds_offset(&Vs[0][0]);

  // Preload Q fragments (A-frag layout: lane r holds Q[row][ka], Q[row][ka+1])
  v2f aq[2][16];
#pragma unroll
  for (int mt = 0; mt < 2; ++mt)
#pragma unroll
    for (int kt = 0; kt < 16; ++kt) {
      const int ka  = kt * 4 + 2 * half;
      const int row = mt * 16 + r;
      aq[mt][kt] = { Q[row * HD_ + ka], Q[row * HD_ + ka + 1] };
    }

  v8f O[2][4] = {};
  float mrow[2][8], lrow[2][8];
#pragma unroll
  for (int mt = 0; mt < 2; ++mt)
#pragma unroll
    for (int v = 0; v < 8; ++v) { mrow[mt][v] = -1e30f; lrow[mt][v] = 0.0f; }

  const int kend = qt * 128 + 128;
  for (int kb = 0; kb < kend; kb += 32) {
    // ---- TDM: stage K and V 32x64 tiles into padded LDS ----
    if (wid == 0) {
      tdm_load_2d_f32(ldsK, Kp + (size_t)kb * HD_, 64, 32, 64, 5);
      __builtin_amdgcn_s_wait_tensorcnt(0);
    } else if (wid == 1) {
      tdm_load_2d_f32(ldsV, Vp + (size_t)kb * HD_, 64, 32, 64, 5);
      __builtin_amdgcn_s_wait_tensorcnt(0);
    }
    __syncthreads();

    if (kb < q0 + 32) {          // wave-uniform: EXEC stays all-1s for WMMA
      // ---- S = Q * K^T  (32x32 scores, 2x2 WMMA subtiles) ----
      v8f S[2][2] = {};
#pragma unroll
      for (int kt = 0; kt < 16; ++kt) {
        const int ka = kt * 4 + 2 * half;
        v2f b0 = { Ks[r][ka],      Ks[r][ka + 1] };
        v2f b1 = { Ks[16 + r][ka], Ks[16 + r][ka + 1] };
        S[0][0] = wmma4(aq[0][kt], b0, S[0][0]);
        S[0][1] = wmma4(aq[0][kt], b1, S[0][1]);
        S[1][0] = wmma4(aq[1][kt], b0, S[1][0]);
        S[1][1] = wmma4(aq[1][kt], b1, S[1][1]);
      }

      // ---- causal mask ----
#pragma unroll
      for (int mt = 0; mt < 2; ++mt)
#pragma unroll
        for (int nt = 0; nt < 2; ++nt)
#pragma unroll
          for (int v = 0; v < 8; ++v) {
            const int qrow = q0 + mt * 16 + half * 8 + v;
            const int kcol = kb + nt * 16 + r;
            if (kcol > qrow) S[mt][nt][v] = -1e30f;
          }

      // ---- online softmax update ----
#pragma unroll
      for (int mt = 0; mt < 2; ++mt)
#pragma unroll
        for (int v = 0; v < 8; ++v) {
          float mx = fmaxf(S[mt][0][v], S[mt][1][v]);
#pragma unroll
          for (int off = 1; off < 16; off <<= 1)
            mx = fmaxf(mx, __shfl_xor(mx, off, 32));
          const float mnew = fmaxf(mrow[mt][v], mx);
          const float corr = __expf(mrow[mt][v] - mnew);
          mrow[mt][v] = mnew;
          const float p0 = __expf(S[mt][0][v] - mnew);
          const float p1 = __expf(S[mt][1][v] - mnew);
          S[mt][0][v] = p0;
          S[mt][1][v] = p1;
          float rs = p0 + p1;
#pragma unroll
          for (int off = 1; off < 16; off <<= 1)
            rs += __shfl_xor(rs, off, 32);
          lrow[mt][v] = lrow[mt][v] * corr + rs;
#pragma unroll
          for (int nt = 0; nt < 4; ++nt) O[mt][nt][v] *= corr;
        }

      // ---- stage P through LDS: C-frag layout -> A-frag layout ----
#pragma unroll
      for (int mt = 0; mt < 2; ++mt)
#pragma unroll
        for (int nt = 0; nt < 2; ++nt)
#pragma unroll
          for (int v = 0; v < 8; ++v)
            Pb[wid][mt * 16 + half * 8 + v][nt * 16 + r] = S[mt][nt][v];
      __builtin_amdgcn_wave_barrier();

      // ---- O += P * V ----
#pragma unroll
      for (int kt = 0; kt < 8; ++kt) {
        const int ka = kt * 4 + 2 * half;
        v2f ap0 = { Pb[wid][r][ka],      Pb[wid][r][ka + 1] };
        v2f ap1 = { Pb[wid][16 + r][ka], Pb[wid][16 + r][ka + 1] };
#pragma unroll
        for (int nt = 0; nt < 4; ++nt) {
          v2f bv = { Vs[ka][nt * 16 + r], Vs[ka + 1][nt * 16 + r] };
          O[0][nt] = wmma4(ap0, bv, O[0][nt]);
          O[1][nt] = wmma4(ap1, bv, O[1][nt]);
        }
      }
    }
    __syncthreads();
  }

  // ---- epilogue: normalize and write y in (B,T,NH*HD) layout ----
#pragma unroll
  for (int mt = 0; mt < 2; ++mt)
#pragma unroll
    for (int v = 0; v < 8; ++v) {
      const float inv = 1.0f / lrow[mt][v];
      const int qrow  = q0 + mt * 16 + half * 8 + v;
#pragma unroll
      for (int nt = 0; nt < 4; ++nt) {
        const float o = O[mt][nt][v] * inv;
        y[((size_t)(b * T_ + qrow)) * D_ + h * HD_ + nt * 16 + r] = o;
      }
    }
}

// ---------------------------------------------------------------------------
extern "C" void kernel_launch(void* const* d_in, const int* in_sizes, int n_in,
                              void* d_out, int out_size, void* d_ws, size_t ws_size,
                              hipStream_t stream) {
  const float* x      = (const float*)d_in[0];
  const float* cosp   = (const float*)d_in[1];
  const float* sinp   = (const float*)d_in[2];
  const float* v0     = (const float*)d_in[3];
  const float* Wqkv   = (const float*)d_in[4];
  const float* Wproj  = (const float*)d_in[5];
  const float* alpha  = (const float*)d_in[6];
  const float* qscale = (const float*)d_in[7];
  float* out = (float*)d_out;

  float* ws  = (float*)d_ws;
  float* qkv = ws;                                   // 4096 * 1536
  float* qs  = qkv + (size_t)B_ * T_ * E_;           // B*NH*T*HD
  float* ks  = qs  + (size_t)B_ * NH_  * T_ * HD_;   // B*NKV*T*HD
  float* vm  = ks  + (size_t)B_ * NKV_ * T_ * HD_;   // B*NH*T*HD
  float* y   = qkv;                                  // reuse dead qkv region

  const int M = B_ * T_;                             // 4096

  // 1) qkv = x @ Wqkv^T
  gemm_xwT<<<dim3(E_ / 64, M / 128), 256, 0, stream>>>(x, Wqkv, qkv, M, E_, D_);

  // 2) RoPE + scale folding + value-residual mix
  rope_prep<<<B_ * T_, 256, 0, stream>>>(qkv, cosp, sinp, v0, alpha, qscale,
                                         qs, ks, vm);

  // 3) causal flash attention -> y (overwrites qkv region)
  attn<<<dim3(T_ / 128, NH_, B_), 128, 0, stream>>>(qs, ks, vm, y);

  // 4) out = y @ Wproj^T
  gemm_xwT<<<dim3(D_ / 64, M / 128), 256, 0, stream>>>(y, Wproj, out, M, D_, D_);

  // 5) second tuple element: pass-through copy of v0
  hipMemcpyAsync(out + (size_t)B_ * T_ * D_, v0,
                 (size_t)B_ * NH_ * T_ * HD_ * sizeof(float),
                 hipMemcpyDeviceToDevice, stream);
}